// ConditionalRandomField_78529182040328
// MI455X (gfx1250) — compile-verified
//
#include <hip/hip_runtime.h>
#include <math.h>

#define TAGS      64
#define START_TAG 62
#define STOP_TAG  63
#define NEGV      (-10000.0f)
#define BSZ       512
#define LSZ       512
#define NB        16            // batches per scan block
#define SCAN_THREADS 128        // 4 wave32

typedef float v2f __attribute__((ext_vector_type(2)));
typedef float v8f __attribute__((ext_vector_type(8)));
typedef int   v4i __attribute__((vector_size(16)));

#if defined(__AMDGCN__) && \
    __has_builtin(__builtin_amdgcn_global_load_async_to_lds_b128) && \
    __has_builtin(__builtin_amdgcn_s_wait_asynccnt)
#define USE_ASYNC 1
#else
#define USE_ASYNC 0
#endif

__device__ __forceinline__ void emit_async_b128(const float* g, float* l) {
#if USE_ASYNC
    __builtin_amdgcn_global_load_async_to_lds_b128(
        (__attribute__((address_space(1))) v4i*)g,
        (__attribute__((address_space(3))) v4i*)l, 0, 0);
#endif
}

// workspace layout (floats)
#define WS_EEXP  0              // 64*64: exp(trans[i][j] - rowmax[i])
#define WS_RMAX  4096           // 64
#define WS_LOGD  4160           // 512
#define WS_SCORE 4672           // 512

// ---------------------------------------------------------------------------
// Kernel 0: rowmax + exp-normalized transition matrix (tiny, one-shot)
// ---------------------------------------------------------------------------
__global__ void crf_prep_kernel(const float* __restrict__ trans,
                                float* __restrict__ eexp,
                                float* __restrict__ rmax) {
    int i = threadIdx.x;
    if (i < TAGS) {
        float m = -3.4e38f;
        for (int j = 0; j < TAGS; ++j) m = fmaxf(m, trans[i * TAGS + j]);
        rmax[i] = m;
        for (int j = 0; j < TAGS; ++j)
            eexp[i * TAGS + j] = __expf(trans[i * TAGS + j] - m);
    }
}

// ---------------------------------------------------------------------------
// Kernel 1: forward scan.  Block = 4 wave32 = NB=16 batches.  Per step:
//   W(64x16) = E'(64x64) x V(64x16) via 16x V_WMMA_F32_16X16X4_F32 per wave,
//   async double-buffered emission tiles, whole mask slab resident in LDS.
// ---------------------------------------------------------------------------
__global__ __launch_bounds__(SCAN_THREADS)
void crf_scan_kernel(const float* __restrict__ inputs,
                     const int*   __restrict__ mask,
                     const float* __restrict__ trans,
                     const float* __restrict__ ws_eexp,
                     const float* __restrict__ ws_rmax,
                     float*       __restrict__ logden) {
    // rows padded to 68 floats -> conflict-free strided access on 64 banks
    __shared__ float sE[TAGS][68];           // E' (resident)
    __shared__ float sAlpha[NB][68];         // forward variables
    __shared__ float sVT[NB][68];            // exp(alpha - m), batch-major
    __shared__ float sEmit[2][NB][68];       // double-buffered emissions
    __shared__ float sMaskAll[NB][LSZ + 4];  // whole mask slab, padded rows
    __shared__ float sRmax[TAGS];
    __shared__ float sM[NB], sS[NB];

    const int tid  = threadIdx.x;
    const int wave = tid >> 5;
    const int lane = tid & 31;
    const int b0   = blockIdx.x * NB;

    for (int idx = tid; idx < TAGS * TAGS; idx += SCAN_THREADS)
        sE[idx >> 6][idx & 63] = ws_eexp[idx];
    for (int i = tid; i < TAGS; i += SCAN_THREADS) sRmax[i] = ws_rmax[i];
    for (int idx = tid; idx < NB * TAGS; idx += SCAN_THREADS) {
        int b = idx >> 6, j = idx & 63;
        sAlpha[b][j] = (j == START_TAG) ? 0.0f : NEGV;
    }
    for (int idx = tid; idx < NB * LSZ; idx += SCAN_THREADS) {
        int bl = idx >> 9, t = idx & (LSZ - 1);
        sMaskAll[bl][t] = (float)mask[(size_t)(b0 + bl) * LSZ + t];
    }
    __syncthreads();

    // ---- hoist A fragments (E' constant across scan) ----
    // f32 16x4 A layout: lanes 0-15: M=lane,K={0,1}; lanes 16-31: M=lane-16,K={2,3}
    const int rowBase = wave << 4;
    const int half    = lane >> 4;
    const int lr      = lane & 15;
    const int rowA    = rowBase + lr;
    v2f afrag[16];
#pragma unroll
    for (int kb = 0; kb < 16; ++kb) {
        afrag[kb].x = sE[rowA][(kb << 2) + (half << 1) + 0];
        afrag[kb].y = sE[rowA][(kb << 2) + (half << 1) + 1];
    }
    float rmaxReg[8];
#pragma unroll
    for (int r = 0; r < 8; ++r) rmaxReg[r] = sRmax[rowBase + r + (half << 3)];

    const int mygrp = tid >> 3;   // batch owned in V-phase
    const int sub   = tid & 7;    // 8 lanes per batch

#if USE_ASYNC
    // prime buffer 0 with t=0 emissions
#pragma unroll
    for (int k = 0; k < 2; ++k) {
        int i4 = tid + k * SCAN_THREADS;
        int bl = i4 >> 4, off = (i4 & 15) << 2;
        emit_async_b128(inputs + ((size_t)(b0 + bl) * LSZ) * TAGS + off,
                        &sEmit[0][bl][off]);
    }
#endif

    int cur = 0;
    for (int t = 0; t < LSZ; ++t) {
        const int nxt = cur ^ 1;
#if USE_ASYNC
        if (t + 1 < LSZ) {      // stream next step's tile while we compute
#pragma unroll
            for (int k = 0; k < 2; ++k) {
                int i4 = tid + k * SCAN_THREADS;
                int bl = i4 >> 4, off = (i4 & 15) << 2;
                emit_async_b128(
                    inputs + ((size_t)(b0 + bl) * LSZ + (t + 1)) * TAGS + off,
                    &sEmit[nxt][bl][off]);
            }
        }
#else
#pragma unroll
        for (int k = 0; k < 2; ++k) {
            int i4 = tid + k * SCAN_THREADS;
            int bl = i4 >> 4, off = (i4 & 15) << 2;
            const float* src = inputs + ((size_t)(b0 + bl) * LSZ + t) * TAGS + off;
            *reinterpret_cast<float4*>(&sEmit[cur][bl][off]) =
                *reinterpret_cast<const float4*>(src);
            if (t + 1 < LSZ) __builtin_prefetch(src + TAGS, 0, 0);
        }
#endif
        // ---- V phase: m_b, V = exp(alpha - m), s_b ----
        float a8[8];
        float mloc = NEGV;
#pragma unroll
        for (int jj = 0; jj < 8; ++jj) {
            a8[jj] = sAlpha[mygrp][(sub << 3) + jj];
            mloc   = fmaxf(mloc, a8[jj]);
        }
#pragma unroll
        for (int off = 4; off >= 1; off >>= 1)
            mloc = fmaxf(mloc, __shfl_xor(mloc, off, 8));
        float vr[8];
        float sloc = 0.0f;
#pragma unroll
        for (int jj = 0; jj < 8; ++jj) {
            vr[jj] = __expf(a8[jj] - mloc);
            sloc  += vr[jj];
        }
        *reinterpret_cast<float4*>(&sVT[mygrp][(sub << 3) + 0]) =
            make_float4(vr[0], vr[1], vr[2], vr[3]);
        *reinterpret_cast<float4*>(&sVT[mygrp][(sub << 3) + 4]) =
            make_float4(vr[4], vr[5], vr[6], vr[7]);
#pragma unroll
        for (int off = 4; off >= 1; off >>= 1)
            sloc += __shfl_xor(sloc, off, 8);
        if (sub == 0) { sM[mygrp] = mloc; sS[mygrp] = sloc; }

#if USE_ASYNC
        if (t + 1 < LSZ) __builtin_amdgcn_s_wait_asynccnt(2);  // t's pair landed
        else             __builtin_amdgcn_s_wait_asynccnt(0);
#endif
        __syncthreads();

        // ---- preload all B fragments, then chain 16 WMMAs (K = 64) ----
        // B value V[k][n] lives at sVT[n][k];  bfrag vgpr0 -> K rows {0,1},
        // vgpr1 -> {2,3}; half-wave picks pair member, N = lane&15.
        v2f bfrag[16];
#pragma unroll
        for (int kb = 0; kb < 16; ++kb) {
            bfrag[kb].x = sVT[lr][(kb << 2) + half];
            bfrag[kb].y = sVT[lr][(kb << 2) + half + 2];
        }
        v8f acc = {};
#pragma unroll
        for (int kb = 0; kb < 16; ++kb)
            acc = __builtin_amdgcn_wmma_f32_16x16x4_f32(
                false, afrag[kb], false, bfrag[kb], (short)0, acc, false, false);

        // ---- epilogue: alpha' = log(W) + rowmax + m_b + emit (mask-aware) ----
        const float mb = sM[lr], sb = sS[lr];
        const float mk = sMaskAll[lr][t];
        const float fallback = __logf(sb) + mb;  // mask==0: logsumexp(alpha)
        const int   iBase    = rowBase + (half << 3);
        float outv[8];
#pragma unroll
        for (int r = 0; r < 8; ++r) {
            float w   = fmaxf(acc[r], 1e-30f);
            float upd = __logf(w) + rmaxReg[r] + mb + sEmit[cur][lr][iBase + r] * mk;
            outv[r]   = (mk != 0.0f) ? upd : fallback;
        }
        *reinterpret_cast<float4*>(&sAlpha[lr][iBase + 0]) =
            make_float4(outv[0], outv[1], outv[2], outv[3]);
        *reinterpret_cast<float4*>(&sAlpha[lr][iBase + 4]) =
            make_float4(outv[4], outv[5], outv[6], outv[7]);
        __syncthreads();
        cur = nxt;
    }

    // ---- terminal: logZ_b = logsumexp_i(alpha[i] + trans[STOP][i]) ----
    if (tid < NB) {
        float m2 = -3.4e38f;
        for (int j = 0; j < TAGS; ++j)
            m2 = fmaxf(m2, sAlpha[tid][j] + trans[STOP_TAG * TAGS + j]);
        float s2 = 0.0f;
        for (int j = 0; j < TAGS; ++j)
            s2 += __expf(sAlpha[tid][j] + trans[STOP_TAG * TAGS + j] - m2);
        logden[b0 + tid] = __logf(s2) + m2;
    }
}

// ---------------------------------------------------------------------------
// Kernel 2: gold-path numerator score per batch (gathers)
// ---------------------------------------------------------------------------
__global__ __launch_bounds__(128)
void crf_score_kernel(const float* __restrict__ inputs,
                      const int*   __restrict__ tags,
                      const int*   __restrict__ mask,
                      const float* __restrict__ trans,
                      float*       __restrict__ score) {
    __shared__ float red[128];
    const int b   = blockIdx.x;
    const int tid = threadIdx.x;
    const int*   tg   = tags + (size_t)b * LSZ;
    const int*   mk   = mask + (size_t)b * LSZ;
    const float* in_b = inputs + (size_t)b * LSZ * TAGS;

    float acc = 0.0f;
    for (int t = tid; t < LSZ; t += 128) {
        int   g = tg[t];
        float m = (float)mk[t];
        acc += in_b[(size_t)t * TAGS + g] * m;
        if (t > 0) acc += trans[g * TAGS + tg[t - 1]] * m;
    }
    red[tid] = acc;
    __syncthreads();
    for (int s = 64; s >= 1; s >>= 1) {
        if (tid < s) red[tid] += red[tid + s];
        __syncthreads();
    }
    if (tid == 0)
        score[b] = red[0] + trans[tg[0] * TAGS + START_TAG]
                          + trans[STOP_TAG * TAGS + tg[LSZ - 1]];
}

// ---------------------------------------------------------------------------
// Kernel 3: loss = sum_b (score[b] - logZ[b])
// ---------------------------------------------------------------------------
__global__ __launch_bounds__(512)
void crf_reduce_kernel(const float* __restrict__ score,
                       const float* __restrict__ logden,
                       float* __restrict__ out) {
    __shared__ float red[512];
    int tid = threadIdx.x;
    red[tid] = score[tid] - logden[tid];
    __syncthreads();
    for (int s = 256; s >= 1; s >>= 1) {
        if (tid < s) red[tid] += red[tid + s];
        __syncthreads();
    }
    if (tid == 0) out[0] = red[0];
}

// ---------------------------------------------------------------------------
extern "C" void kernel_launch(void* const* d_in, const int* in_sizes, int n_in,
                              void* d_out, int out_size, void* d_ws, size_t ws_size,
                              hipStream_t stream) {
    const float* inputs = (const float*)d_in[0];   // (512,512,64) f32
    const int*   tags   = (const int*)  d_in[1];   // (512,512) i32
    const int*   mask   = (const int*)  d_in[2];   // (512,512) i32
    const float* trans  = (const float*)d_in[3];   // (64,64) f32
    float* ws  = (float*)d_ws;
    float* out = (float*)d_out;

    float* eexp   = ws + WS_EEXP;
    float* rmax   = ws + WS_RMAX;
    float* logden = ws + WS_LOGD;
    float* score  = ws + WS_SCORE;

    crf_prep_kernel<<<1, 64, 0, stream>>>(trans, eexp, rmax);
    crf_scan_kernel<<<BSZ / NB, SCAN_THREADS, 0, stream>>>(
        inputs, mask, trans, eexp, rmax, logden);
    crf_score_kernel<<<BSZ, 128, 0, stream>>>(inputs, tags, mask, trans, score);
    crf_reduce_kernel<<<1, 512, 0, stream>>>(score, logden, out);
}